// DetectionLoss_4827543241340
// MI455X (gfx1250) — compile-verified
//
#include <hip/hip_runtime.h>

// ---- problem constants (from reference setup) ----
#define NGT      24
#define THRESHV  0.35f
#define NEGPOS   7
#define VAR0     0.1f
#define VAR1     0.2f

typedef float v2f __attribute__((ext_vector_type(2)));
typedef float v8f __attribute__((ext_vector_type(8)));

// Wave32 sum reduction routed through V_WMMA_F32_16X16X4_F32.
// A = ones(16x4), B holds one lane value per lane (other slot 0).
// D[m,n] = sum_K B[K,n]  (identical for every row m), so
// sum over all 32 lanes of d[0] covers every column twice:
//   0.5 * shfl-sum(d[0]) == sum of all B entries == sum over lanes of v.
// This identity is robust to the exact K/N slot assignment.
__device__ __forceinline__ float wave_sum32(float v) {
#if __has_builtin(__builtin_amdgcn_wmma_f32_16x16x4_f32)
  v2f a; a[0] = 1.0f; a[1] = 1.0f;
  v2f b; b[0] = v;    b[1] = 0.0f;
  v8f c = {0.f, 0.f, 0.f, 0.f, 0.f, 0.f, 0.f, 0.f};
  c = __builtin_amdgcn_wmma_f32_16x16x4_f32(false, a, false, b, (short)0, c,
                                            false, false);
  float s = c[0];
#pragma unroll
  for (int off = 16; off > 0; off >>= 1) s += __shfl_xor(s, off, 32);
  return 0.5f * s;
#else
  float s = v;
#pragma unroll
  for (int off = 16; off > 0; off >>= 1) s += __shfl_xor(s, off, 32);
  return s;
#endif
}

__device__ __forceinline__ float smooth_l1(float x) {
  float d = fabsf(x);
  return (d < 1.0f) ? 0.5f * d * d : d - 0.5f;
}

// ---------------- kernel 1: zero the cross-call state ----------------
__global__ void k_init(unsigned long long* bestkey, int* num_pos, float* acc,
                       int B, int G) {
  int i = blockIdx.x * blockDim.x + threadIdx.x;
  if (i < B * G) bestkey[i] = 0ull;
  if (i < B) num_pos[i] = 0;
  if (i < 3) acc[i] = 0.0f;
}

// ---------------- kernel 2: per-prior best-truth + per-truth best-prior ----
__global__ void k_match(const float4* __restrict__ priors,   // [P] cx,cy,w,h
                        const float4* __restrict__ gt,       // [B,G] x1,y1,x2,y2
                        float* __restrict__ bto, int* __restrict__ bti,
                        unsigned long long* __restrict__ bestkey,
                        int P, int G) {
  const int b   = blockIdx.y;
  const int p   = blockIdx.x * blockDim.x + threadIdx.x;
  const int tid = threadIdx.x;

  __shared__ float4 sgt[NGT];
  __shared__ unsigned long long skey[NGT];
  if (tid < G) {
    sgt[tid]  = gt[(size_t)b * G + tid];
    skey[tid] = 0ull;
  }
  __syncthreads();

  if (p < P) {
    float4 pr = priors[p];
    float px1 = pr.x - 0.5f * pr.z, py1 = pr.y - 0.5f * pr.w;
    float px2 = pr.x + 0.5f * pr.z, py2 = pr.y + 0.5f * pr.w;
    float area_p = (px2 - px1) * (py2 - py1);

    float best = -1.0f;
    int   bi   = 0;
#pragma unroll 4
    for (int g = 0; g < NGT; ++g) {
      float4 t  = sgt[g];
      float iw  = fmaxf(fminf(t.z, px2) - fmaxf(t.x, px1), 0.0f);
      float ih  = fmaxf(fminf(t.w, py2) - fmaxf(t.y, py1), 0.0f);
      float inter  = iw * ih;
      float area_t = (t.z - t.x) * (t.w - t.y);
      float iou = inter / (area_t + area_p - inter);
      if (iou > best) { best = iou; bi = g; }
      // packed argmax key: larger iou wins; ties -> smaller p (argmax semantics)
      unsigned long long key =
          ((unsigned long long)__float_as_uint(iou) << 32) |
          (unsigned long long)(0x7FFFFFFFu - (unsigned)p);
      atomicMax(&skey[g], key);  // LDS u64 atomic (ds_max_u64)
    }
    bto[(size_t)b * P + p] = best;
    bti[(size_t)b * P + p] = bi;
  }
  __syncthreads();
  if (tid < G) atomicMax(&bestkey[(size_t)b * G + tid], skey[tid]);
}

// ---------------- kernel 3: ensure each GT keeps its best prior ------------
__global__ void k_override(const unsigned long long* __restrict__ bestkey,
                           float* __restrict__ bto, int* __restrict__ bti,
                           int P, int G) {
  int b = blockIdx.x;
  if (threadIdx.x == 0) {
    for (int g = 0; g < G; ++g) {  // sequential: last g wins, like .at[].set
      unsigned long long key = bestkey[(size_t)b * G + g];
      unsigned p = 0x7FFFFFFFu - (unsigned)(key & 0xFFFFFFFFull);
      if (p < (unsigned)P) {
        bto[(size_t)b * P + p] = 2.0f;
        bti[(size_t)b * P + p] = g;
      }
    }
  }
}

// ---------------- kernel 4: loc loss, CE, mining array ---------------------
__global__ void k_loss(const float4* __restrict__ loc,    // [B,P]
                       const float2* __restrict__ conf,   // [B,P]
                       const float4* __restrict__ gt,     // [B,G]
                       const int* __restrict__ labels,    // [B,G]
                       const float4* __restrict__ priors, // [P]
                       const float* __restrict__ bto,
                       const int* __restrict__ bti,
                       float* __restrict__ mine,
                       int* __restrict__ num_pos,
                       float* __restrict__ acc,           // [0]=loss_l [1]=ce_pos
                       int P, int G) {
  const int b = blockIdx.y;
  const int p = blockIdx.x * blockDim.x + threadIdx.x;
  const bool valid = (p < P);

  float lossl = 0.0f, cepos = 0.0f;
  bool pos = false;

  if (valid) {
    size_t idx = (size_t)b * P + p;
    // pull the streaming operands toward L2/L0 early (global_prefetch_b8)
    __builtin_prefetch(&conf[idx], 0, 1);
    __builtin_prefetch(&loc[idx], 0, 1);

    float o = bto[idx];
    int   g = bti[idx];
    int  lb = labels[(size_t)b * G + g];
    int tgt = (o < THRESHV) ? 0 : lb;
    pos = (tgt > 0);

    float2 cd = conf[idx];
    float m   = fmaxf(cd.x, cd.y);
    float lse = m + logf(expf(cd.x - m) + expf(cd.y - m));
    float gold = (tgt == 1) ? cd.y : cd.x;
    float ce = lse - gold;
    mine[idx] = pos ? 0.0f : ce;

    if (pos) {
      cepos = ce;
      float4 t  = gt[(size_t)b * G + g];
      float4 pr = priors[p];
      float gcx = ((t.x + t.z) * 0.5f - pr.x) / (VAR0 * pr.z);
      float gcy = ((t.y + t.w) * 0.5f - pr.y) / (VAR0 * pr.w);
      float gw  = logf((t.z - t.x) / pr.z) / VAR1;
      float gh  = logf((t.w - t.y) / pr.w) / VAR1;
      float4 l = loc[idx];
      lossl = smooth_l1(l.x - gcx) + smooth_l1(l.y - gcy) +
              smooth_l1(l.z - gw)  + smooth_l1(l.w - gh);
    }
  }

  // EXEC is all-ones here (full 256-thread blocks, reconverged) -> WMMA legal
  float wl = wave_sum32(lossl);
  float wc = wave_sum32(cepos);
  unsigned long long bm = __ballot(pos ? 1 : 0);
  if ((threadIdx.x & 31) == 0) {
    if (wl != 0.0f) atomicAdd(&acc[0], wl);
    if (wc != 0.0f) atomicAdd(&acc[1], wc);
    int cnt = __popcll(bm);
    if (cnt) atomicAdd(&num_pos[b], cnt);
  }
}

// ---------------- kernel 5: per-batch top-k negative sum (radix select) ----
__global__ void k_radix(const float* __restrict__ mine,
                        const int* __restrict__ num_pos,
                        float* __restrict__ acc, int P) {
  const int b   = blockIdx.x;
  const int tid = threadIdx.x;

  long long nn = (long long)NEGPOS * (long long)num_pos[b];
  int num_neg = (int)(nn < (long long)(P - 1) ? nn : (long long)(P - 1));
  if (num_neg <= 0) return;  // uniform across block

  __shared__ unsigned int hist[256];
  __shared__ unsigned int s_prefix;
  __shared__ int s_k;
  __shared__ float ssum[256];
  __shared__ int   scnt[256];

  unsigned prefix = 0u;
  int k = num_neg;
  const float* mb = mine + (size_t)b * P;

  for (int pass = 0; pass < 4; ++pass) {
    int shift = 24 - 8 * pass;
    hist[tid] = 0u;
    __syncthreads();
    unsigned himask = (pass == 0) ? 0u : (0xFFFFFFFFu << (shift + 8));
    for (int p = tid; p < P; p += 256) {
      if (p + 2048 < P) __builtin_prefetch(&mb[p + 2048], 0, 1);
      unsigned u = __float_as_uint(mb[p]);  // mine >= 0 -> bits ordered
      if ((u & himask) == prefix) atomicAdd(&hist[(u >> shift) & 255u], 1u);
    }
    __syncthreads();
    if (tid == 0) {
      unsigned above = 0u;
      int sel = 0;
      for (int bin = 255; bin >= 0; --bin) {
        unsigned c = hist[bin];
        if (above + c >= (unsigned)k) { sel = bin; k -= (int)above; break; }
        above += c;
      }
      s_prefix = prefix | ((unsigned)sel << shift);
      s_k = k;
    }
    __syncthreads();
    prefix = s_prefix;
    k = s_k;
    __syncthreads();
  }

  float thresh = __uint_as_float(prefix);  // exact k-th largest value
  float lsum = 0.0f;
  int lcnt = 0;
  for (int p = tid; p < P; p += 256) {
    if (p + 2048 < P) __builtin_prefetch(&mb[p + 2048], 0, 1);
    float v = mb[p];
    if (v > thresh) { lsum += v; ++lcnt; }
  }
  ssum[tid] = lsum;
  scnt[tid] = lcnt;
  __syncthreads();
  for (int s = 128; s > 0; s >>= 1) {
    if (tid < s) { ssum[tid] += ssum[tid + s]; scnt[tid] += scnt[tid + s]; }
    __syncthreads();
  }
  if (tid == 0) {
    // ties at thresh are interchangeable: exact sum regardless of which selected
    float contrib = ssum[0] + (float)(num_neg - scnt[0]) * thresh;
    atomicAdd(&acc[2], contrib);
  }
}

// ---------------- kernel 6: combine --------------------------------------
__global__ void k_finalize(const int* __restrict__ num_pos,
                           const float* __restrict__ acc,
                           float* __restrict__ out, int B) {
  if (blockIdx.x == 0 && threadIdx.x == 0) {
    int tot = 0;
    for (int i = 0; i < B; ++i) tot += num_pos[i];
    float N = fmaxf((float)tot, 1.0f);
    out[0] = (2.0f * acc[0] + acc[1] + acc[2]) / N;  // LOC_WEIGHT*loss_l + loss_c
  }
}

extern "C" void kernel_launch(void* const* d_in, const int* in_sizes, int n_in,
                              void* d_out, int out_size, void* d_ws,
                              size_t ws_size, hipStream_t stream) {
  const float* loc  = (const float*)d_in[0];  // [B,P,4]
  const float* conf = (const float*)d_in[1];  // [B,P,2]
  const float* gt   = (const float*)d_in[2];  // [B,G,4]
  const int*   lab  = (const int*)d_in[3];    // [B,G]
  const float* pri  = (const float*)d_in[4];  // [P,4]

  const int G = NGT;
  const int B = in_sizes[3] / G;
  const int P = in_sizes[4] / 4;
  const size_t BP = (size_t)B * (size_t)P;

  char* ws = (char*)d_ws;
  float* bto = (float*)ws;
  int*   bti = (int*)(ws + BP * sizeof(float));
  float* mine = (float*)(ws + 2 * BP * sizeof(float));
  unsigned long long* bestkey =
      (unsigned long long*)(ws + 3 * BP * sizeof(float));
  int* num_pos =
      (int*)(ws + 3 * BP * sizeof(float) + (size_t)B * G * sizeof(unsigned long long));
  float* acc = (float*)((char*)num_pos + (size_t)B * sizeof(int));

  dim3 blk(256);
  dim3 gbp((P + 255) / 256, B);

  k_init<<<(B * G + 255) / 256, blk, 0, stream>>>(bestkey, num_pos, acc, B, G);
  k_match<<<gbp, blk, 0, stream>>>((const float4*)pri, (const float4*)gt, bto,
                                   bti, bestkey, P, G);
  k_override<<<B, 32, 0, stream>>>(bestkey, bto, bti, P, G);
  k_loss<<<gbp, blk, 0, stream>>>((const float4*)loc, (const float2*)conf,
                                  (const float4*)gt, lab, (const float4*)pri,
                                  bto, bti, mine, num_pos, acc, P, G);
  k_radix<<<B, 256, 0, stream>>>(mine, num_pos, acc, P);
  k_finalize<<<1, 32, 0, stream>>>(num_pos, acc, (float*)d_out, B);
}